// YoloV1Loss_12927851561239
// MI455X (gfx1250) — compile-verified
//
#include <hip/hip_runtime.h>

// YOLOv1 loss for MI455X (gfx1250, wave32).
// Memory-bound streaming reduction: LDS-staged coalesced loads, per-cell math,
// shfl + LDS + v_wmma_f32_16x16x4_f32 block reduction, two-stage deterministic
// final reduce (no float atomics -> bit-identical across graph replays).

typedef __attribute__((ext_vector_type(2))) float v2f;
typedef __attribute__((ext_vector_type(8))) float v8f;

#define CELLS_PER_BLOCK 256
#define FLOATS_PER_CELL 30
#define STAGE_FLOATS (CELLS_PER_BLOCK * FLOATS_PER_CELL)   // 7680
#define STAGE_F4     (STAGE_FLOATS / 4)                    // 1920

// Block-wide reduction of 5 per-thread accumulators.
// Requires blockDim.x == 256 (8 waves of 32). Thread 0 writes dst[l] = total_l * scale_l.
__device__ __forceinline__ void block_reduce5_wmma(float v0, float v1, float v2,
                                                   float v3, float v4,
                                                   float* sRed,   // >= 64 floats
                                                   float* dst,
                                                   float s0, float s1, float s2,
                                                   float s3, float s4) {
    float v[5] = {v0, v1, v2, v3, v4};
    // intra-wave tree (wave32)
    #pragma unroll
    for (int off = 16; off > 0; off >>= 1) {
        #pragma unroll
        for (int l = 0; l < 5; ++l) v[l] += __shfl_down(v[l], off);
    }
    const int lane = threadIdx.x & 31;
    const int wave = threadIdx.x >> 5;
    if (lane == 0) {
        #pragma unroll
        for (int l = 0; l < 5; ++l) sRed[wave * 8 + l] = v[l];
    }
    __syncthreads();
    if (wave == 0) {
        // A is 16x4 fp32: lane L(0..15) holds A[L][0],A[L][1]; lane L+16 holds A[L][2],A[L][3].
        // Row m (m<5) = loss m; A[m][k] = wavePartial[2k][m] + wavePartial[2k+1][m].
        const int m  = lane & 15;
        const int kb = (lane >> 4) * 2;   // 0 -> k={0,1}, 2 -> k={2,3}
        v2f a; a.x = 0.0f; a.y = 0.0f;
        if (m < 5) {
            a.x = sRed[(2 * kb + 0) * 8 + m] + sRed[(2 * kb + 1) * 8 + m];
            a.y = sRed[(2 * kb + 2) * 8 + m] + sRed[(2 * kb + 3) * 8 + m];
        }
        v2f b; b.x = 1.0f; b.y = 1.0f;    // ones B: D[m][n] = sum_k A[m][k]
        v8f c = {};
        // EXEC is full for wave 0 here (branch is wave-uniform, divergence above reconverged).
        c = __builtin_amdgcn_wmma_f32_16x16x4_f32(false, a, false, b, (short)0, c,
                                                  false, false);
        if (lane == 0) {
            // D VGPR v, lane 0 => D[v][0] = total for loss v.
            dst[0] = c[0] * s0;
            dst[1] = c[1] * s1;
            dst[2] = c[2] * s2;
            dst[3] = c[3] * s3;
            dst[4] = c[4] * s4;
        }
    }
}

__global__ __launch_bounds__(256) void yolo_v1_loss_main(
        const float* __restrict__ pred, const float* __restrict__ targ,
        float* __restrict__ partials) {
    __shared__ __align__(16) float sP[STAGE_FLOATS];
    __shared__ __align__(16) float sT[STAGE_FLOATS];
    __shared__ float sRed[64];

    const int tid = threadIdx.x;
    const size_t blockBase = (size_t)blockIdx.x * STAGE_FLOATS;

    // Fully-coalesced b128 staging of 256 cells x 30 floats for both tensors.
    const float4* gP = (const float4*)(pred + blockBase);
    const float4* gT = (const float4*)(targ + blockBase);
    float4* lP = (float4*)sP;
    float4* lT = (float4*)sT;
    for (int idx = tid; idx < STAGE_F4; idx += 256) {
        lP[idx] = gP[idx];
        lT[idx] = gT[idx];
    }
    __syncthreads();

    const float* cp = sP + tid * FLOATS_PER_CELL;
    const float* ct = sT + tid * FLOATS_PER_CELL;

    const float t4  = ct[4];
    const float coo = (t4 > 0.0f) ? 1.0f : 0.0f;
    const float noo = 1.0f - coo;

    // Target box -> xyxy (faithful to reference: divide by S, subtract for areas).
    const float tcx = ct[0] / 7.0f, tcy = ct[1] / 7.0f;
    const float tw  = ct[2],        th  = ct[3];
    const float tx0 = tcx - 0.5f * tw, ty0 = tcy - 0.5f * th;
    const float tx1 = tcx + 0.5f * tw, ty1 = tcy + 0.5f * th;
    const float areaT = (tx1 - tx0) * (ty1 - ty0);

    float iou[2], conf[2], bx[2], by[2], bw[2], bh[2];
    #pragma unroll
    for (int bxi = 0; bxi < 2; ++bxi) {
        const float* pb = cp + bxi * 5;
        bx[bxi] = pb[0]; by[bxi] = pb[1];
        bw[bxi] = pb[2]; bh[bxi] = pb[3];
        conf[bxi] = pb[4];
        const float cx = pb[0] / 7.0f, cy = pb[1] / 7.0f;
        const float x0 = cx - 0.5f * pb[2], y0 = cy - 0.5f * pb[3];
        const float x1 = cx + 0.5f * pb[2], y1 = cy + 0.5f * pb[3];
        const float lx = fmaxf(x0, tx0), ly = fmaxf(y0, ty0);
        const float rx = fminf(x1, tx1), ry = fminf(y1, ty1);
        const float wx = fmaxf(rx - lx, 0.0f), wy = fmaxf(ry - ly, 0.0f);
        const float inter = wx * wy;
        const float areaP = (x1 - x0) * (y1 - y0);
        iou[bxi] = inter / (areaP + areaT - inter);
    }

    // jnp.argmax tie-break: first index wins -> box 1 only if strictly greater.
    const int   r       = (iou[1] > iou[0]) ? 1 : 0;
    const float max_iou = fmaxf(iou[0], iou[1]);
    const float cr = conf[r], cn = conf[1 - r];

    const float dcr = cr - max_iou;
    float acc_contain = coo * dcr * dcr;

    const float dx = bx[r] - ct[0];
    const float dy = by[r] - ct[1];
    const float dw = sqrtf(bw[r]) - sqrtf(ct[2]);
    const float dh = sqrtf(bh[r]) - sqrtf(ct[3]);
    float acc_loc = coo * (dx * dx + dy * dy + dw * dw + dh * dh);

    float acc_notc = coo * cn * cn;

    const float d4 = cp[4] - ct[4];
    const float d9 = cp[9] - ct[9];
    float acc_noo = noo * (d4 * d4 + d9 * d9);

    float cls = 0.0f;
    #pragma unroll
    for (int k = 10; k < 30; ++k) {
        const float d = cp[k] - ct[k];
        cls += d * d;
    }
    float acc_cls = coo * cls;

    __syncthreads();  // staging LDS reads complete before reduction phase
    block_reduce5_wmma(acc_loc, acc_contain, acc_notc, acc_noo, acc_cls,
                       sRed, partials + (size_t)blockIdx.x * 5,
                       1.0f, 1.0f, 1.0f, 1.0f, 1.0f);
}

__global__ __launch_bounds__(256) void yolo_v1_loss_finalize(
        const float* __restrict__ partials, float* __restrict__ out,
        int nblocks, float invN) {
    __shared__ float sRed[64];
    float v[5] = {0.0f, 0.0f, 0.0f, 0.0f, 0.0f};
    // Fixed strided order -> deterministic accumulation across replays.
    for (int i = threadIdx.x; i < nblocks; i += 256) {
        #pragma unroll
        for (int l = 0; l < 5; ++l) v[l] += partials[(size_t)i * 5 + l];
    }
    block_reduce5_wmma(v[0], v[1], v[2], v[3], v[4], sRed, out,
                       5.0f * invN,   // L_COORD * loc / N
                       2.0f * invN,   // 2 * contain / N
                       1.0f * invN,   // not_contain / N
                       0.5f * invN,   // L_NOOBJ * nooobj / N
                       1.0f * invN);  // class / N
}

extern "C" void kernel_launch(void* const* d_in, const int* in_sizes, int n_in,
                              void* d_out, int out_size, void* d_ws, size_t ws_size,
                              hipStream_t stream) {
    const float* pred = (const float*)d_in[0];
    const float* targ = (const float*)d_in[1];
    float* out      = (float*)d_out;
    float* partials = (float*)d_ws;   // nblocks * 5 floats

    const int total   = in_sizes[0];                 // N * 7 * 7 * 30
    const int ncells  = total / FLOATS_PER_CELL;     // N * 49
    const int nblocks = ncells / CELLS_PER_BLOCK;    // 3136 for N=16384
    const int N       = ncells / 49;
    const float invN  = 1.0f / (float)N;

    yolo_v1_loss_main<<<nblocks, 256, 0, stream>>>(pred, targ, partials);
    yolo_v1_loss_finalize<<<1, 256, 0, stream>>>(partials, out, nblocks, invN);
}